// VQVAE_18760417149134
// MI455X (gfx1250) — compile-verified
//
#include <hip/hip_runtime.h>
#include <hip/hip_bf16.h>

typedef __attribute__((ext_vector_type(2))) float v2f;
typedef __attribute__((ext_vector_type(8))) float v8f;

// Problem constants (from reference): B=256, C=512, K=1024, E=64
#define C_DIM 512
#define K_DIM 1024
#define E_DIM 64
#define CE    32768            // C*E
#define CK    524288           // C*K
#define Z_ELEMS 8388608        // B*C*E
#define OH_BASE 16777216       // 2*B*C*E

// ---------------------------------------------------------------------------
// Kernel 1: d2[c*K + k] = sum_e dictionary[c][k][e]^2   (C*K = 524288 values)
// ---------------------------------------------------------------------------
__global__ __launch_bounds__(256) void vqvae_d2_kernel(const float* __restrict__ dict,
                                                       float* __restrict__ d2) {
    int i = blockIdx.x * blockDim.x + threadIdx.x;   // i = c*K + k
    if (i >= CK) return;
    const float* p = dict + (size_t)i * E_DIM;
    float s = 0.0f;
#pragma unroll
    for (int e = 0; e < E_DIM; e += 4) {
        float4 v = *(const float4*)(p + e);
        s += v.x * v.x + v.y * v.y + v.z * v.z + v.w * v.w;
    }
    d2[i] = s;
}

// ---------------------------------------------------------------------------
// Kernel 2: per (c, b-tile of 16) workgroup:
//   - WMMA f32 16x16x4 dot products over E=64 (16 k-steps) for all K=1024
//   - argmin_k (d2 - 2*dot)  [m2 is constant in k -> dropped]
//   - write z, z_embed, one_hot slabs for this tile
// ---------------------------------------------------------------------------
#define NWAVES 8

__global__ __launch_bounds__(256) void vqvae_main_kernel(const float* __restrict__ mu,
                                                         const float* __restrict__ dict,
                                                         const float* __restrict__ d2,
                                                         float* __restrict__ out) {
    const int c  = blockIdx.x;        // 0..511
    const int b0 = blockIdx.y * 16;   // batch tile start

    const int tid  = threadIdx.x;
    const int lane = tid & 31;
    const int wave = tid >> 5;
    const int half = lane >> 4;       // 0: lanes 0-15, 1: lanes 16-31
    const int ln   = lane & 15;

    __shared__ float s_val[NWAVES][16];
    __shared__ int   s_idx[NWAVES][16];
    __shared__ int   s_best[16];

    // ---- preload A fragments (mu tile, 16 rows x 64 e) -------------------
    // 32-bit A 16x4 layout: lanes 0-15 hold K=0 (v0), K=1 (v1);
    //                       lanes 16-31 hold K=2 (v0), K=3 (v1).
    v2f afrag[16];
    const float* mub = mu + (size_t)(b0 + ln) * CE + (size_t)c * E_DIM;
#pragma unroll
    for (int s = 0; s < 16; ++s) {
        const int eoff = 4 * s + 2 * half;
        afrag[s] = *(const v2f*)(mub + eoff);
    }

    const float* dictc = dict + (size_t)c * K_DIM * E_DIM;
    const float* d2c   = d2 + (size_t)c * K_DIM;

    float bv[8];
    int   bk[8];
#pragma unroll
    for (int r = 0; r < 8; ++r) { bv[r] = 3.402823466e38f; bk[r] = 0; }

    // ---- 8 k-tiles per wave, 16 WMMA steps each --------------------------
    for (int t = 0; t < 8; ++t) {
        const int k0 = (wave * 8 + t) * 16;
        v8f acc = {};
#pragma unroll
        for (int s = 0; s < 16; ++s) {
            const int eoff = 4 * s + 2 * half;
            // B 4x16: lane ln supplies column n=ln; v0 = K=eoff, v1 = K=eoff+1
            v2f bfrag = *(const v2f*)(dictc + (size_t)(k0 + ln) * E_DIM + eoff);
            acc = __builtin_amdgcn_wmma_f32_16x16x4_f32(
                false, afrag[s], false, bfrag, (short)0, acc, false, false);
        }
        const int   kk  = k0 + ln;
        const float dk2 = d2c[kk];
#pragma unroll
        for (int r = 0; r < 8; ++r) {
            // C/D layout: vgpr r, lanes 0-15 -> (M=r, N=ln); lanes 16-31 -> (M=r+8, N=ln)
            const float dist = dk2 - 2.0f * acc[r];
            if (dist < bv[r] || (dist == bv[r] && kk < bk[r])) { bv[r] = dist; bk[r] = kk; }
        }
    }

    // ---- reduce argmin across the 16 lanes of each half (wave32 shfl) ----
#pragma unroll
    for (int r = 0; r < 8; ++r) {
        float v = bv[r];
        int   k = bk[r];
#pragma unroll
        for (int m = 1; m < 16; m <<= 1) {
            const float ov = __shfl_xor(v, m, 32);
            const int   ok = __shfl_xor(k, m, 32);
            if (ov < v || (ov == v && ok < k)) { v = ov; k = ok; }
        }
        bv[r] = v; bk[r] = k;
    }
    if (ln == 0) {
        // lane 0 holds rows b=0..7 (half 0), lane 16 rows b=8..15 (half 1)
#pragma unroll
        for (int r = 0; r < 8; ++r) {
            s_val[wave][half * 8 + r] = bv[r];
            s_idx[wave][half * 8 + r] = bk[r];
        }
    }
    __syncthreads();

    // ---- reduce across waves (each wave covered a distinct k range) ------
    if (tid < 16) {
        float v = s_val[0][tid];
        int   k = s_idx[0][tid];
#pragma unroll
        for (int w = 1; w < NWAVES; ++w) {
            const float ov = s_val[w][tid];
            const int   ok = s_idx[w][tid];
            if (ov < v || (ov == v && ok < k)) { v = ov; k = ok; }
        }
        s_best[tid] = k;
    }
    __syncthreads();

    // ---- outputs ---------------------------------------------------------
    // z and z_embed: 16 rows x 64 floats; each thread handles 4 floats.
    {
        const int b  = tid >> 4;           // 0..15
        const int e0 = (tid & 15) * 4;     // 0..60
        const int kb = s_best[b];
        const float4 dv = *(const float4*)(dictc + (size_t)kb * E_DIM + e0);
        const size_t moff = (size_t)(b0 + b) * CE + (size_t)c * E_DIM + e0;
        const float4 mv = *(const float4*)(mu + moff);
        float4 z;
        z.x = mv.x + (dv.x - mv.x);
        z.y = mv.y + (dv.y - mv.y);
        z.z = mv.z + (dv.z - mv.z);
        z.w = mv.w + (dv.w - mv.w);
        *(float4*)(out + moff)           = z;   // z
        *(float4*)(out + Z_ELEMS + moff) = dv;  // z_embed
    }
    // one_hot: 16 rows x 1024 k; each thread owns a 64-wide k strip of one row.
    {
        const int b     = tid >> 4;
        const int kbase = (tid & 15) * 64;
        const int kb    = s_best[b];
        float* oh = out + OH_BASE + (size_t)(b0 + b) * CK + (size_t)c * K_DIM + kbase;
#pragma unroll
        for (int j = 0; j < 64; j += 4) {
            float4 v;
            v.x = (kbase + j + 0 == kb) ? 1.0f : 0.0f;
            v.y = (kbase + j + 1 == kb) ? 1.0f : 0.0f;
            v.z = (kbase + j + 2 == kb) ? 1.0f : 0.0f;
            v.w = (kbase + j + 3 == kb) ? 1.0f : 0.0f;
            *(float4*)(oh + j) = v;
        }
    }
}

extern "C" void kernel_launch(void* const* d_in, const int* in_sizes, int n_in,
                              void* d_out, int out_size, void* d_ws, size_t ws_size,
                              hipStream_t stream) {
    (void)in_sizes; (void)n_in; (void)out_size; (void)ws_size;
    const float* mu   = (const float*)d_in[0];   // (256, 32768) fp32
    const float* dict = (const float*)d_in[1];   // (512, 1024, 64) fp32
    float* out = (float*)d_out;                  // z | z_embed | one_hot
    float* d2  = (float*)d_ws;                   // C*K floats = 2 MB scratch

    // Pass 1: code norms d2[c][k]
    vqvae_d2_kernel<<<dim3(CK / 256), dim3(256), 0, stream>>>(dict, d2);

    // Pass 2: WMMA dots + argmin + outputs
    vqvae_main_kernel<<<dim3(C_DIM, 16), dim3(256), 0, stream>>>(mu, dict, d2, out);
}